// TeacherSeedEncoder_40699110097148
// MI455X (gfx1250) — compile-verified
//
#include <hip/hip_runtime.h>
#include <hip/hip_bf16.h>

typedef __attribute__((ext_vector_type(2))) float v2f;
typedef __attribute__((ext_vector_type(8))) float v8f;

#define B_SZ        64
#define N_OBJ       16384
#define Q_LIDAR     2048
#define NUM_CLASSES 23
#define IN_DIM      33      // 9 boxes + 1 score + 23 one-hot
#define FPAD        36      // K padded to multiple of 4 for 16x16x4 WMMA
#define MD          512
#define HSTRIDE     516     // 512 + 4 pad: kills LDS bank conflicts on column reads
#define ROWS_PER_WG 128
#define NEG_INF     (-1e30f)

// ---- CDNA5 async global->LDS copy (ASYNCcnt-tracked) via inline asm --------
// ISA 15.18.3: GLOBAL_LOAD_ASYNC_TO_LDS_B128  vdst(LDS byte addr), vaddr, off
// Generic->LDS address: aperture rule (ISA 10.2): LDS address is the low 32 bits of it,
// so truncating the generic pointer gives the LDS byte address for VDST.
__device__ __forceinline__ void cp16_g2lds(const float* __restrict__ g, float* l) {
#if defined(__gfx1250__)
    unsigned lofs = (unsigned)(unsigned long long)l;
    asm volatile("global_load_async_to_lds_b128 %0, %1, off"
                 :: "v"(lofs), "v"(g) : "memory");
#else
    *(float4*)l = *(const float4*)g;
#endif
}

__device__ __forceinline__ void wait_async_copies() {
#if defined(__gfx1250__)
#if __has_builtin(__builtin_amdgcn_s_wait_asynccnt)
    __builtin_amdgcn_s_wait_asynccnt(0);
#else
    asm volatile("s_wait_asynccnt 0" ::: "memory");
#endif
#endif
}

// ---------------------------------------------------------------------------
// Kernel 1: exact per-batch top-k via full bitonic sort of 64-bit keys in LDS.
// key = (monotonic-uint(score_masked) << 32) | ~index  -> descending sort gives
// jax.lax.top_k order (desc value, asc index on ties).  128 KB LDS per WG.
// ---------------------------------------------------------------------------
__global__ __launch_bounds__(1024)
void topk_sort_kernel(const float* __restrict__ scores,
                      const int*   __restrict__ valid,
                      int*         __restrict__ sel_idx)
{
    extern __shared__ unsigned long long keys[];   // N_OBJ entries
    const int b = blockIdx.x;
    const float* s = scores + (size_t)b * N_OBJ;
    const int*   v = valid  + (size_t)b * N_OBJ;

    for (int i = threadIdx.x; i < N_OBJ; i += blockDim.x) {
        float sc = v[i] ? s[i] : NEG_INF;
        unsigned u = __float_as_uint(sc);
        u = (u & 0x80000000u) ? ~u : (u | 0x80000000u);   // order-preserving map
        keys[i] = ((unsigned long long)u << 32) | (unsigned)(~(unsigned)i);
    }
    __syncthreads();

    for (int k = 2; k <= N_OBJ; k <<= 1) {
        for (int j = k >> 1; j > 0; j >>= 1) {
            for (int i = threadIdx.x; i < N_OBJ; i += blockDim.x) {
                int ixj = i ^ j;
                if (ixj > i) {
                    unsigned long long a = keys[i], c = keys[ixj];
                    bool up = ((i & k) == 0);      // descending comparator
                    if (up ? (a < c) : (a > c)) { keys[i] = c; keys[ixj] = a; }
                }
            }
            __syncthreads();
        }
    }

    for (int q = threadIdx.x; q < Q_LIDAR; q += blockDim.x)
        sel_idx[(size_t)b * Q_LIDAR + q] =
            (int)(~(unsigned)(keys[q] & 0xFFFFFFFFull));
}

// ---------------------------------------------------------------------------
// Kernel 2: gather + feature build + GEMM1(ReLU) + GEMM2 + masked outputs.
// 256 threads = 8 waves; each wave owns a private 16-row strip of H, so no
// inter-wave barrier is needed between GEMM1 and GEMM2.  GEMM2's w2 column
// slice (512x16 = 32 KB) is staged cooperatively in LDS (async copy) and
// shared by all 8 waves -> 8x less L2 traffic than per-wave streaming.
// All matrix math via V_WMMA_F32_16X16X4_F32 (exact fp32; compute is free
// vs the 268 MB output-store roofline, so no precision sacrifice needed).
// ---------------------------------------------------------------------------
__global__ __launch_bounds__(256)
void seed_encoder_kernel(const float* __restrict__ boxes,
                         const float* __restrict__ scores,
                         const float* __restrict__ w1,
                         const float* __restrict__ b1,
                         const float* __restrict__ w2,
                         const float* __restrict__ b2,
                         const int*   __restrict__ labels,
                         const int*   __restrict__ valid,
                         const int*   __restrict__ sel_idx,
                         float* __restrict__ qout,
                         float* __restrict__ refs_out,
                         float* __restrict__ sc_out)
{
    extern __shared__ float lds[];
    float* feat   = lds;                                   // 128 * 36
    float* Hlds   = feat   + ROWS_PER_WG * FPAD;           // 128 * 516
    float* svalid = Hlds   + ROWS_PER_WG * HSTRIDE;        // 128
    float* w2s    = svalid + ROWS_PER_WG;                  // 512 * 16 (32 KB)

    const int b  = blockIdx.y;
    const int r0 = blockIdx.x * ROWS_PER_WG;
    const int t  = threadIdx.x;

    // Warm L2 for w2 (1 MB = 8192 x 128B lines) -> global_prefetch_b8
    for (int i = t; i < 8192; i += 256)
        __builtin_prefetch((const char*)w2 + (size_t)i * 128, 0, 1);

    // ---- Gather + feature construction (threads 0..127, one row each) ----
    if (t < ROWS_PER_WG) {
        const int q   = r0 + t;
        const int src = sel_idx[(size_t)b * Q_LIDAR + q];
        const float* bx = boxes + ((size_t)b * N_OBJ + src) * 9;
        const float  sc = scores[(size_t)b * N_OBJ + src];
        const int    lb = labels[(size_t)b * N_OBJ + src];
        const int    vl = valid [(size_t)b * N_OBJ + src];

        float* f = feat + t * FPAD;
        #pragma unroll
        for (int c = 0; c < FPAD; ++c) f[c] = 0.0f;
        float bx0 = bx[0], bx1 = bx[1], bx2 = bx[2];
        f[0] = bx0; f[1] = bx1; f[2] = bx2;
        f[3] = bx[3]; f[4] = bx[4]; f[5] = bx[5];
        f[6] = bx[6]; f[7] = bx[7]; f[8] = bx[8];
        f[9] = sc;
        if (lb >= 0 && lb < NUM_CLASSES) f[10 + lb] = 1.0f;

        const float m = vl ? 1.0f : 0.0f;
        svalid[t] = m;

        size_t qo = (size_t)b * Q_LIDAR + q;
        refs_out[qo * 3 + 0] = m * bx0;
        refs_out[qo * 3 + 1] = m * bx1;
        refs_out[qo * 3 + 2] = m * bx2;
        sc_out[qo]           = m * sc;
    }
    __syncthreads();

    // ---- WMMA layouts (wave32) ----
    // A 16x4 f32 : lane l -> M = l%16, VGPR r -> K = r + 2*(l/16)
    // B 4x16 f32 : lane l -> N = l%16, VGPR r -> K = r + 2*(l/16)
    // C/D 16x16  : lane l -> N = l%16, VGPR r -> M = r + 8*(l/16)
    const int w    = t >> 5;
    const int l    = t & 31;
    const int half = l >> 4;
    const int n16  = l & 15;
    const int koff = 2 * half;

    float* Fw = feat + (w * 16) * FPAD;     // this wave's 16 feature rows
    float* Hw = Hlds + (w * 16) * HSTRIDE;  // this wave's 16 hidden rows

    // ---------------- GEMM1: H = relu(F @ w1 + b1) ----------------
    // w1 is only 66 KB and read ~once per wave -> stream from L2 directly.
    for (int nt = 0; nt < MD / 16; ++nt) {
        const float bv = b1[nt * 16 + n16];
        v8f acc = {bv, bv, bv, bv, bv, bv, bv, bv};
        #pragma unroll
        for (int kk = 0; kk < FPAD; kk += 4) {
            const int k0 = kk + koff;
            v2f a, bb;
            a[0]  = Fw[n16 * FPAD + k0];
            a[1]  = Fw[n16 * FPAD + k0 + 1];
            bb[0] = (k0     < IN_DIM) ? w1[(size_t)k0       * MD + nt * 16 + n16] : 0.0f;
            bb[1] = (k0 + 1 < IN_DIM) ? w1[(size_t)(k0 + 1) * MD + nt * 16 + n16] : 0.0f;
            acc = __builtin_amdgcn_wmma_f32_16x16x4_f32(
                false, a, false, bb, (short)0, acc, false, false);
        }
        #pragma unroll
        for (int r = 0; r < 8; ++r) {
            float h = acc[r];
            h = h > 0.0f ? h : 0.0f;                       // ReLU
            Hw[(r + 8 * half) * HSTRIDE + nt * 16 + n16] = h;
        }
    }
    // No barrier needed here: each wave consumes only its own H strip, and
    // the w2s buffer is disjoint from feat/H.

    // ---------------- GEMM2: P = H @ w2 + b2, masked store ----------------
    for (int nt = 0; nt < MD / 16; ++nt) {
        // Cooperative stage of w2[:, nt*16 .. nt*16+15] (512x16 fp32) into LDS.
        // 2048 float4 transfers / 256 threads = 8 async 16B copies per thread.
        #pragma unroll
        for (int it = 0; it < 8; ++it) {
            const int idx = t + it * 256;        // float4 index 0..2047
            const int k   = idx >> 2;            // w2 row   0..511
            const int c   = (idx & 3) << 2;      // col chunk 0,4,8,12
            cp16_g2lds(w2 + (size_t)k * MD + nt * 16 + c, w2s + k * 16 + c);
        }
        wait_async_copies();
        __syncthreads();                         // slice visible to all waves

        const float bv = b2[nt * 16 + n16];
        v8f acc = {bv, bv, bv, bv, bv, bv, bv, bv};
        #pragma unroll 4
        for (int kk = 0; kk < MD; kk += 4) {
            const int k0 = kk + koff;
            v2f a, bb;
            a[0]  = Hw[n16 * HSTRIDE + k0];
            a[1]  = Hw[n16 * HSTRIDE + k0 + 1];
            bb[0] = w2s[k0 * 16 + n16];          // conflict-free: 16 lanes span
            bb[1] = w2s[(k0 + 1) * 16 + n16];    // 16 consecutive banks
            acc = __builtin_amdgcn_wmma_f32_16x16x4_f32(
                false, a, false, bb, (short)0, acc, false, false);
        }
        #pragma unroll
        for (int r = 0; r < 8; ++r) {
            const int row = r + 8 * half;                  // row within 16-strip
            const int q   = r0 + w * 16 + row;
            const float m = svalid[w * 16 + row];
            qout[((size_t)b * Q_LIDAR + q) * MD + nt * 16 + n16] = m * acc[r];
        }
        __syncthreads();                         // all done before next overwrite
    }
}

// ---------------------------------------------------------------------------
extern "C" void kernel_launch(void* const* d_in, const int* in_sizes, int n_in,
                              void* d_out, int out_size, void* d_ws, size_t ws_size,
                              hipStream_t stream) {
    const float* boxes  = (const float*)d_in[0];   // [B,N,9]
    const float* scores = (const float*)d_in[1];   // [B,N]
    const float* w1     = (const float*)d_in[2];   // [33,512]
    const float* b1     = (const float*)d_in[3];   // [512]
    const float* w2     = (const float*)d_in[4];   // [512,512]
    const float* b2     = (const float*)d_in[5];   // [512]
    const int*   labels = (const int*)d_in[6];     // [B,N]
    const int*   valid  = (const int*)d_in[7];     // [B,N]

    float* qout  = (float*)d_out;                               // [B,Q,512]
    float* refs  = qout + (size_t)B_SZ * Q_LIDAR * MD;          // [B,Q,3]
    float* scout = refs + (size_t)B_SZ * Q_LIDAR * 3;           // [B,Q]

    int* sel = (int*)d_ws;                                      // [B,Q] indices

    // Allow >64KB dynamic LDS (CDNA5 WGP has 320KB). Not a stream op; graph-safe.
    size_t shm1 = (size_t)N_OBJ * sizeof(unsigned long long);            // 128 KB
    size_t shm2 = (size_t)(ROWS_PER_WG * FPAD + ROWS_PER_WG * HSTRIDE +
                           ROWS_PER_WG + MD * 16) * sizeof(float);       // ~308.5 KB
    (void)hipFuncSetAttribute((const void*)topk_sort_kernel,
        hipFuncAttributeMaxDynamicSharedMemorySize, (int)shm1);
    (void)hipFuncSetAttribute((const void*)seed_encoder_kernel,
        hipFuncAttributeMaxDynamicSharedMemorySize, (int)shm2);

    topk_sort_kernel<<<B_SZ, 1024, shm1, stream>>>(scores, valid, sel);

    dim3 grid(Q_LIDAR / ROWS_PER_WG, B_SZ);   // 16 x 64 workgroups
    seed_encoder_kernel<<<grid, 256, shm2, stream>>>(
        boxes, scores, w1, b1, w2, b2, labels, valid, sel, qout, refs, scout);
}